// InstanceWeightedCrossEntropyLoss_45887430590783
// MI455X (gfx1250) — compile-verified
//
#include <hip/hip_runtime.h>

// InstanceWeightedCrossEntropyLoss for B=32, C=2, H=W=1024.
// Stage 1: stream 384 MB once (NT b128 loads), per-block partial (s0, s1, n1).
// Stage 2: WMMA(ones x partials) reduction tree per sample + final mean.

typedef float  v4f  __attribute__((ext_vector_type(4)));
typedef int    v4i  __attribute__((ext_vector_type(4)));
typedef float  v2f  __attribute__((ext_vector_type(2)));
typedef float  v8f  __attribute__((ext_vector_type(8)));

#define HW_PIX   (1024 * 1024)
#define NSAMP    32
#define BPS      256          // stage-1 blocks per sample
#define TPB      256          // threads per block (8 wave32s)
#define VITER    4            // 256 blocks * 256 thr * 4 iter * 4 px = 1M px

// ---------------------------------------------------------------- stage 1 ---
__global__ __launch_bounds__(TPB)
void iwce_stage1(const float* __restrict__ logits,
                 const int*   __restrict__ targets,
                 float*       __restrict__ ws)
{
    const int b = blockIdx.x / BPS;            // sample
    const int j = blockIdx.x - b * BPS;        // block within sample

    const float* __restrict__ x0p = logits + (size_t)b * (2ull * HW_PIX);
    const float* __restrict__ x1p = x0p + HW_PIX;
    const int*   __restrict__ tp  = targets + (size_t)b * HW_PIX;

    float s0 = 0.f, s1 = 0.f, n1 = 0.f;
    const int vbase = j * TPB + (int)threadIdx.x;      // float4 index

#pragma unroll
    for (int i = 0; i < VITER; ++i) {
        const int    v  = vbase + i * (BPS * TPB);
        const size_t px = (size_t)v * 4;
        // Read-once 384MB stream: non-temporal so we don't churn the 192MB L2.
        v4f a = __builtin_nontemporal_load((const v4f*)(x0p + px));
        v4f c = __builtin_nontemporal_load((const v4f*)(x1p + px));
        v4i t = __builtin_nontemporal_load((const v4i*)(tp  + px));
#pragma unroll
        for (int e = 0; e < 4; ++e) {
            const float xa  = a[e];
            const float xc  = c[e];
            const float m   = fmaxf(xa, xc);
            const float tail = log1pf(__expf(-fabsf(xa - xc)));
            const bool  one = (t[e] != 0);
            const float nll = (m - (one ? xc : xa)) + tail;
            s0 += one ? 0.f : nll;
            s1 += one ? nll : 0.f;
            n1 += one ? 1.f : 0.f;
        }
    }

    // wave32 reduction
#pragma unroll
    for (int off = 16; off > 0; off >>= 1) {
        s0 += __shfl_down(s0, off, 32);
        s1 += __shfl_down(s1, off, 32);
        n1 += __shfl_down(n1, off, 32);
    }

    __shared__ float sm[3][TPB / 32];
    const int wave = threadIdx.x >> 5;
    if ((threadIdx.x & 31) == 0) {
        sm[0][wave] = s0; sm[1][wave] = s1; sm[2][wave] = n1;
    }
    __syncthreads();
    if (threadIdx.x == 0) {
        float t0 = 0.f, t1 = 0.f, t2 = 0.f;
#pragma unroll
        for (int w = 0; w < TPB / 32; ++w) {
            t0 += sm[0][w]; t1 += sm[1][w]; t2 += sm[2][w];
        }
        // ws layout per sample: [0,256)=s0 plane, [256,512)=s1, [512,768)=n1
        float* dst = ws + (size_t)b * (3 * BPS);
        dst[j]           = t0;
        dst[BPS + j]     = t1;
        dst[2 * BPS + j] = t2;
    }
}

// ---------------------------------------------------------------- stage 2 ---
// One wave per sample (32 waves). Sum of 256 partials per quantity via a
// chain of 4x V_WMMA_F32_16X16X4_F32 with A = ones(16x4):
//   D[m][n] = sum_k B[k][n] + C[m][n]  -> every lane's c[0] = colsum[lane&15]
// B(4x16) element (k,n) lives at lane = n + 16*(k>>1), vgpr = k&1.
__global__ __launch_bounds__(1024)
void iwce_stage2(const float* __restrict__ ws, float* __restrict__ out)
{
    const int s    = threadIdx.x >> 5;     // sample handled by this wave
    const int lane = threadIdx.x & 31;
    const float* __restrict__ base = ws + (size_t)s * (3 * BPS);

    const int lo = (lane & 16) ? 32 : 0;   // k>=2 half sits in lanes 16..31
    const int n  = lane & 15;

    v2f ones; ones[0] = 1.f; ones[1] = 1.f;
    v8f c0 = {0.f, 0.f, 0.f, 0.f, 0.f, 0.f, 0.f, 0.f};
    v8f c1 = c0;
    v8f c2 = c0;

#pragma unroll
    for (int chunk = 0; chunk < 4; ++chunk) {
        const int i0 = chunk * 64 + lo + n;
        v2f b0; b0[0] = base[i0];            b0[1] = base[i0 + 16];
        v2f b1; b1[0] = base[256 + i0];      b1[1] = base[256 + i0 + 16];
        v2f b2; b2[0] = base[512 + i0];      b2[1] = base[512 + i0 + 16];
        c0 = __builtin_amdgcn_wmma_f32_16x16x4_f32(false, ones, false, b0,
                                                   (short)0, c0, false, false);
        c1 = __builtin_amdgcn_wmma_f32_16x16x4_f32(false, ones, false, b1,
                                                   (short)0, c1, false, false);
        c2 = __builtin_amdgcn_wmma_f32_16x16x4_f32(false, ones, false, b2,
                                                   (short)0, c2, false, false);
    }

    // c[0] now holds colsum[lane&15]; fold the 16 columns.
    float S0 = c0[0], S1 = c1[0], C1 = c2[0];
#pragma unroll
    for (int m = 8; m >= 1; m >>= 1) {
        S0 += __shfl_xor(S0, m, 32);
        S1 += __shfl_xor(S1, m, 32);
        C1 += __shfl_xor(C1, m, 32);
    }

    // Weights: w_c = 1 - cnt_c/HW; blend = 0.4*w + 0.6*mean, mean == 0.5 (C=2)
    const float cnt1 = C1;
    const float cnt0 = (float)HW_PIX - cnt1;
    const float inv  = 1.f / (float)HW_PIX;
    const float w0   = 0.4f * (1.f - cnt0 * inv) + 0.3f;
    const float w1   = 0.4f * (1.f - cnt1 * inv) + 0.3f;
    const float per  = (w0 * S0 + w1 * S1) / (w0 * cnt0 + w1 * cnt1);

    __shared__ float ps[NSAMP];
    if (lane == 0) ps[s] = per;
    __syncthreads();

    if (threadIdx.x < 32) {
        float v = ps[threadIdx.x];
#pragma unroll
        for (int m = 16; m >= 1; m >>= 1) v += __shfl_xor(v, m, 32);
        if (threadIdx.x == 0) out[0] = v * (1.f / (float)NSAMP);
    }
}

// ----------------------------------------------------------------- launch ---
extern "C" void kernel_launch(void* const* d_in, const int* in_sizes, int n_in,
                              void* d_out, int out_size, void* d_ws, size_t ws_size,
                              hipStream_t stream)
{
    const float* logits  = (const float*)d_in[0];
    const int*   targets = (const int*)d_in[1];   // jax default: int64 -> int32
    float*       ws      = (float*)d_ws;          // 32*768*4 = 96 KB used
    float*       out     = (float*)d_out;

    iwce_stage1<<<dim3(NSAMP * BPS), dim3(TPB), 0, stream>>>(logits, targets, ws);
    iwce_stage2<<<dim3(1), dim3(1024), 0, stream>>>(ws, out);
}